// SVDPPModel_13503377179007
// MI455X (gfx1250) — compile-verified
//
#include <hip/hip_runtime.h>
#include <math.h>

typedef float v2f __attribute__((ext_vector_type(2)));
typedef float v8f __attribute__((ext_vector_type(8)));

#define GLOBAL_BIAS_F 3.5f
#define LAMDA_F 0.5f

// ---------------- K1: item out-degree histogram (float counts) ----------------
__global__ __launch_bounds__(256) void k_count(const int* __restrict__ src,
                                               float* __restrict__ cnt, int n) {
    int e = blockIdx.x * blockDim.x + threadIdx.x;
    if (e < n) atomicAdd(&cnt[src[e]], 1.0f);
}

// ---------------- K2: in-place reciprocal (guard empty bins) ----------------
__global__ __launch_bounds__(256) void k_inv(float* __restrict__ cnt, int n) {
    int i = blockIdx.x * blockDim.x + threadIdx.x;
    if (i < n) {
        float c = cnt[i];
        cnt[i] = (c > 0.0f) ? (1.0f / c) : 0.0f;
    }
}

// ---------------- K3: scatter-add normalized messages into h_user ----------------
// 16 threads per edge, each handles 4 consecutive floats (float4 gather, 4 f32 atomics).
__global__ __launch_bounds__(256) void k_scatter(const int* __restrict__ src,
                                                 const int* __restrict__ dst,
                                                 const float* __restrict__ yw_item,
                                                 const float* __restrict__ inv,
                                                 float* __restrict__ h_user,
                                                 int n_edges) {
    long t = (long)blockIdx.x * blockDim.x + threadIdx.x;
    int e = (int)(t >> 4);
    if (e >= n_edges) return;
    int j = ((int)t & 15) * 4;
    int s = src[e];
    int d = dst[e];
    float w = inv[s];
    float4 y = *(const float4*)(yw_item + (size_t)s * 64 + j);
    float* h = h_user + (size_t)d * 64 + j;
    atomicAdd(h + 0, y.x * w);
    atomicAdd(h + 1, y.y * w);
    atomicAdd(h + 2, y.z * w);
    atomicAdd(h + 3, y.w * w);
}

// ---------------- K4: grid-stride sum of squares -> one accumulator slot ----------------
__global__ __launch_bounds__(256) void k_sumsq(const float* __restrict__ x, long n,
                                               float* __restrict__ acc) {
    __shared__ float sm[256];
    long stride = (long)gridDim.x * blockDim.x;
    float s = 0.0f;
    for (long i = (long)blockIdx.x * blockDim.x + threadIdx.x; i < n; i += stride) {
        float v = x[i];
        s += v * v;
    }
    sm[threadIdx.x] = s;
    __syncthreads();
    for (int o = 128; o > 0; o >>= 1) {
        if ((int)threadIdx.x < o) sm[threadIdx.x] += sm[threadIdx.x + o];
        __syncthreads();
    }
    if (threadIdx.x == 0) atomicAdd(acc, sm[0]);
}

// ---------------- K5: finalize reg_loss ----------------
__global__ void k_reg(const float* __restrict__ acc, float* __restrict__ out_reg,
                      float m_iu, float m_uj) {
    if (blockIdx.x == 0 && threadIdx.x == 0) {
        *out_reg = LAMDA_F * (m_iu * sqrtf(acc[0])    // bias_user
                            + m_uj * sqrtf(acc[1])    // bias_item
                            + m_iu * sqrtf(acc[2])    // pq_user
                            + m_uj * sqrtf(acc[3])    // pq_item
                            + m_uj * sqrtf(acc[4]));  // yw_item
    }
}

// ---------------- K6: WMMA edge scoring ----------------
// One wave32 handles 16 edges. C = A(16x64 of h_user rows) x B(64x16 = pq_item rows^T),
// accumulated via 16 chained V_WMMA_F32_16X16X4_F32. Diagonal of C = the 16 dot products.
// Fragment addressing (per ISA 16x4 A / 4x16 B f32 layouts): lane L supplies a float2 from
// row idx[L&15] at column 4k + 2*(L>>4) for BOTH the A and B fragments.
__global__ __launch_bounds__(256) void k_score_wmma(const float* __restrict__ h_user,
                                                    const float* __restrict__ pq_item,
                                                    const float* __restrict__ bias_user,
                                                    const float* __restrict__ bias_item,
                                                    const int* __restrict__ eu,
                                                    const int* __restrict__ ei,
                                                    float* __restrict__ out,
                                                    int n_tiles) {
    int wave = blockIdx.x * (blockDim.x >> 5) + ((int)threadIdx.x >> 5);
    if (wave >= n_tiles) return;  // wave-uniform exit: EXEC stays all-ones for WMMA
    int lane = (int)threadIdx.x & 31;
    int el = lane & 15;
    int hv = lane >> 4;
    int base = wave * 16;

    int u = eu[base + el];
    int it = ei[base + el];

    const float* ha = h_user + (size_t)u * 64 + 2 * hv;
    const float* pb = pq_item + (size_t)it * 64 + 2 * hv;

    v8f acc = {0.f, 0.f, 0.f, 0.f, 0.f, 0.f, 0.f, 0.f};
#pragma unroll
    for (int k = 0; k < 16; ++k) {
        v2f a = *(const v2f*)(ha + 4 * k);
        v2f b = *(const v2f*)(pb + 4 * k);
        acc = __builtin_amdgcn_wmma_f32_16x16x4_f32(
            /*neg_a=*/false, a, /*neg_b=*/false, b,
            /*c_mod=*/(short)0, acc, /*reuse_a=*/false, /*reuse_b=*/false);
    }

    // Diagonal extraction: m=0..7 -> lane m, vgpr m ; m=8..15 -> lane m+16, vgpr m-8.
    bool active = (lane < 8) || (lane >= 24);
    int m = (lane < 8) ? lane : (lane - 16);
    int s = lane & 7;
    float diag = (s == 0) ? acc[0]
               : (s == 1) ? acc[1]
               : (s == 2) ? acc[2]
               : (s == 3) ? acc[3]
               : (s == 4) ? acc[4]
               : (s == 5) ? acc[5]
               : (s == 6) ? acc[6]
                          : acc[7];
    if (active) {
        // this lane's (u, it) correspond exactly to edge m
        out[base + m] = diag + bias_user[u] + bias_item[it] + GLOBAL_BIAS_F;
    }
}

// ---------------- K7: scalar tail (E_PRED % 16 != 0 safety net) ----------------
__global__ __launch_bounds__(256) void k_score_tail(const float* __restrict__ h_user,
                                                    const float* __restrict__ pq_item,
                                                    const float* __restrict__ bu,
                                                    const float* __restrict__ bi,
                                                    const int* __restrict__ eu,
                                                    const int* __restrict__ ei,
                                                    float* __restrict__ out,
                                                    int start, int n) {
    int e = start + blockIdx.x * blockDim.x + threadIdx.x;
    if (e >= n) return;
    int u = eu[e], it = ei[e];
    const float* a = h_user + (size_t)u * 64;
    const float* b = pq_item + (size_t)it * 64;
    float s = 0.0f;
#pragma unroll
    for (int d = 0; d < 64; ++d) s += a[d] * b[d];
    out[e] = s + bu[u] + bi[it] + GLOBAL_BIAS_F;
}

extern "C" void kernel_launch(void* const* d_in, const int* in_sizes, int n_in,
                              void* d_out, int out_size, void* d_ws, size_t ws_size,
                              hipStream_t stream) {
    const float* pq_user   = (const float*)d_in[0];
    const float* pq_item   = (const float*)d_in[1];
    /* yw_user d_in[2]: unused by the reference computation */
    const float* yw_item   = (const float*)d_in[3];
    const float* bias_user = (const float*)d_in[4];
    const float* bias_item = (const float*)d_in[5];
    const int*   rb_src    = (const int*)d_in[6];
    const int*   rb_dst    = (const int*)d_in[7];
    /* rate_src d_in[8]: mean(I_u) = E_RATE/U exactly, derived analytically */
    const int*   pos_u     = (const int*)d_in[9];
    const int*   pos_i     = (const int*)d_in[10];
    const int*   neg_u     = (const int*)d_in[11];
    const int*   neg_i     = (const int*)d_in[12];

    const int U      = in_sizes[4];   // bias_user flat = U
    const int I      = in_sizes[5];   // bias_item flat = I
    const int E_RB   = in_sizes[6];
    const int E_RATE = in_sizes[8];
    const int E_PRED = in_sizes[9];
    const int D      = 64;

    // workspace layout
    float* cnt   = (float*)d_ws;                      // I floats (counts -> reciprocals)
    float* hus   = cnt + (((size_t)I + 63) & ~63ull); // U*D floats: h_user
    float* accum = hus + (size_t)U * D;               // 5 floats: sum-of-squares slots

    float* out_pos = (float*)d_out;
    float* out_neg = out_pos + E_PRED;
    float* out_reg = out_pos + 2 * (size_t)E_PRED;

    // zero counts + accumulators; h_user starts as pq_user (then messages atomically added)
    hipMemsetAsync(cnt, 0, (size_t)I * sizeof(float), stream);
    hipMemsetAsync(accum, 0, 5 * sizeof(float), stream);
    hipMemcpyAsync(hus, pq_user, (size_t)U * D * sizeof(float),
                   hipMemcpyDeviceToDevice, stream);

    // degree histogram + reciprocal
    k_count<<<(E_RB + 255) / 256, 256, 0, stream>>>(rb_src, cnt, E_RB);
    k_inv<<<(I + 255) / 256, 256, 0, stream>>>(cnt, I);

    // message scatter: E_RB * 16 threads
    {
        long total = (long)E_RB * 16;
        unsigned blocks = (unsigned)((total + 255) / 256);
        k_scatter<<<blocks, 256, 0, stream>>>(rb_src, rb_dst, yw_item, cnt, hus, E_RB);
    }

    // Frobenius-norm sum-of-squares (5 tensors)
    {
        struct { const float* p; long n; int slot; } jobs[5] = {
            {bias_user, (long)U,     0},
            {bias_item, (long)I,     1},
            {pq_user,   (long)U * D, 2},
            {pq_item,   (long)I * D, 3},
            {yw_item,   (long)I * D, 4},
        };
        for (int j = 0; j < 5; ++j) {
            long nb = (jobs[j].n + 255) / 256;
            unsigned blocks = (unsigned)(nb < 2048 ? nb : 2048);
            if (blocks == 0) blocks = 1;
            k_sumsq<<<blocks, 256, 0, stream>>>(jobs[j].p, jobs[j].n, accum + jobs[j].slot);
        }
    }

    // reg_loss with exact analytic means
    float m_iu = (float)E_RATE / (float)U;  // mean(I_u)
    float m_uj = (float)E_RB / (float)I;    // mean(U_j)
    k_reg<<<1, 32, 0, stream>>>(accum, out_reg, m_iu, m_uj);

    // WMMA edge scoring: 16 edges per wave, 8 waves per block
    {
        int n_tiles = E_PRED / 16;
        if (n_tiles > 0) {
            unsigned blocks = (unsigned)((n_tiles + 7) / 8);
            k_score_wmma<<<blocks, 256, 0, stream>>>(hus, pq_item, bias_user, bias_item,
                                                     pos_u, pos_i, out_pos, n_tiles);
            k_score_wmma<<<blocks, 256, 0, stream>>>(hus, pq_item, bias_user, bias_item,
                                                     neg_u, neg_i, out_neg, n_tiles);
        }
        int tail_start = n_tiles * 16;
        if (tail_start < E_PRED) {
            int rem = E_PRED - tail_start;
            unsigned blocks = (unsigned)((rem + 255) / 256);
            k_score_tail<<<blocks, 256, 0, stream>>>(hus, pq_item, bias_user, bias_item,
                                                     pos_u, pos_i, out_pos, tail_start, E_PRED);
            k_score_tail<<<blocks, 256, 0, stream>>>(hus, pq_item, bias_user, bias_item,
                                                     neg_u, neg_i, out_neg, tail_start, E_PRED);
        }
    }
}